// Tensor_product_67190468379307
// MI455X (gfx1250) — compile-verified
//
#include <hip/hip_runtime.h>
#include <stdint.h>

#define LMAX 3
#define NK   156          // total output components
#define NCT  64           // (edge,channel) pairs per block == threads per block
#define ROWF 164          // LDS row stride in floats: 656B, 16B-aligned, odd*4 -> benign banking
#define MAXT 1600

typedef float v4f __attribute__((ext_vector_type(4)));
typedef int   v4i_ls __attribute__((vector_size(16)));   // matches builtin param pointee

// ---------------- CDNA5 async Global<->LDS path ----------------
#if __has_builtin(__builtin_amdgcn_global_load_async_to_lds_b128) && \
    __has_builtin(__builtin_amdgcn_global_store_async_from_lds_b128) && \
    __has_builtin(__builtin_amdgcn_s_wait_asynccnt)
#define USE_ASYNC 1
#else
#define USE_ASYNC 0
#endif

#define AS1 __attribute__((address_space(1)))
#define AS3 __attribute__((address_space(3)))

__device__ __forceinline__ void async_g2l_b128(const void* g, const void* l) {
#if USE_ASYNC
  // builtin signature (probe-confirmed): (v4i AS1* gaddr, v4i AS3* lds, Ii off, Ii cpol)
  __builtin_amdgcn_global_load_async_to_lds_b128(
      (AS1 v4i_ls*)(uintptr_t)g,
      (AS3 v4i_ls*)(uint32_t)(uintptr_t)l, 0, 0);
#else
  (void)g; (void)l;
#endif
}

__device__ __forceinline__ void async_l2g_b128_nt(void* g, const void* l) {
#if USE_ASYNC
  // cpol = 1 -> TH_STORE_NT: 998 MB write-once stream, keep it out of L2.
  __builtin_amdgcn_global_store_async_from_lds_b128(
      (AS1 v4i_ls*)(uintptr_t)g,
      (AS3 v4i_ls*)(uint32_t)(uintptr_t)l, 0, 1);
#else
  (void)g; (void)l;
#endif
}

__device__ __forceinline__ void wait_async0() {
#if USE_ASYNC
  __builtin_amdgcn_s_wait_asynccnt(0);
#endif
}

// ---------------- compile-time Wigner 3j / CG term table ----------------
constexpr double FAC[11] = {1.,1.,2.,6.,24.,120.,720.,5040.,40320.,362880.,3628800.};

constexpr double csqrt(double x) {
  if (x <= 0.0) return 0.0;
  double g = x > 1.0 ? x : 1.0;
  for (int i = 0; i < 32; ++i) g = 0.5 * (g + x / g);
  return g;
}

constexpr double w3j(int j1, int j2, int j3, int m1, int m2, int m3) {
  if (m1 + m2 + m3 != 0) return 0.0;
  int lo = j1 > j2 ? j1 - j2 : j2 - j1;
  if (j3 < lo || j3 > j1 + j2) return 0.0;
  int t1 = j2 - m1 - j3, t2 = j1 + m2 - j3;
  int t3 = j1 + j2 - j3, t4 = j1 - m1, t5 = j2 + m2;
  int tmin = 0; if (t1 > tmin) tmin = t1; if (t2 > tmin) tmin = t2;
  int tmax = t3; if (t4 < tmax) tmax = t4; if (t5 < tmax) tmax = t5;
  if (tmin > tmax) return 0.0;
  double s = 0.0;
  for (int t = tmin; t <= tmax; ++t) {
    double d = FAC[t] * FAC[t - t1] * FAC[t - t2] * FAC[t3 - t] * FAC[t4 - t] * FAC[t5 - t];
    s += (t & 1) ? -1.0 / d : 1.0 / d;
  }
  double p2 = FAC[j1 + j2 - j3] * FAC[j1 - j2 + j3] * FAC[-j1 + j2 + j3] / FAC[j1 + j2 + j3 + 1]
            * FAC[j1 + m1] * FAC[j1 - m1] * FAC[j2 + m2] * FAC[j2 - m2] * FAC[j3 + m3] * FAC[j3 - m3];
  int par = (((j1 - j2 - m3) % 2) + 2) % 2;
  return (par ? -1.0 : 1.0) * csqrt(p2) * s;
}

struct Table {
  int nk;
  int nterms;
  int koff[NK + 1];            // terms for output k live in [koff[k], koff[k+1])
  float coef[MAXT];
  unsigned char prd[MAXT];     // i*7 + j  (index into the 49 products)
};

constexpr Table build_table() {
  Table T{};
  int k = 0, n = 0;
  for (int l1 = 0; l1 <= LMAX; ++l1)
    for (int l2 = 0; l2 <= LMAX; ++l2) {
      int l3lo = l1 > l2 ? l1 - l2 : l2 - l1;
      int l3hi = (l1 + l2) < LMAX ? (l1 + l2) : LMAX;
      for (int l3 = l3lo; l3 <= l3hi; ++l3) {
        for (int m3 = -l3; m3 <= l3; ++m3) {
          T.koff[k] = n;
          for (int m1 = -l1; m1 <= l1; ++m1) {
            int m2 = -m1 - m3;                  // m1+m2+m3 == 0
            if (m2 < -l2 || m2 > l2) continue;
            float c = (float)w3j(l1, l2, l3, m1, m2, m3);
            if (c != 0.0f) {
              T.coef[n] = c;
              T.prd[n] = (unsigned char)((m1 + l1) * 7 + (m2 + l2));
              ++n;
            }
          }
          ++k;
        }
      }
    }
  T.koff[k] = n;
  T.nk = k;
  T.nterms = n;
  return T;
}

constexpr Table TT = build_table();
static_assert(TT.nk == NK, "output dimension mismatch vs reference");
static_assert(TT.nterms <= MAXT, "term table overflow");

// ---------------- kernel ----------------
// One thread per (edge,channel) pair. Sparse CG contraction fully unrolled with
// immediate coefficients; inputs arrive via async b128 DMA into LDS; results are
// staged in LDS and pushed out with async b128 LDS->global NT stores (coalesced,
// no VGPR round-trip). 998 MB output stream dominates the roofline (~47 us).
__global__ __launch_bounds__(NCT) void tp_cg_kernel(const float* __restrict__ x,
                                                    const float* __restrict__ y,
                                                    float* __restrict__ out,
                                                    int nc_total) {
  __shared__ alignas(16) float so[NCT * ROWF];
#if USE_ASYNC
  __shared__ alignas(16) float si[2 * NCT * 7];  // [0,448): x tile, [448,896): y tile
#endif

  const int t   = threadIdx.x;
  const int nc0 = blockIdx.x * NCT;
  const bool full = (nc0 + NCT) <= nc_total;     // uniform per block

  float xs[7], ys[7];

#if USE_ASYNC
  if (full) {
    // Per-block tile: 64*7 floats = 1792 B per array, 16B-aligned chunks.
    const char* xb = (const char*)x + (size_t)nc0 * 28;
    const char* yb = (const char*)y + (size_t)nc0 * 28;
    for (int c = t; c < 112; c += NCT) async_g2l_b128(xb + c * 16, &si[c * 4]);
    for (int c = t; c < 112; c += NCT) async_g2l_b128(yb + c * 16, &si[448 + c * 4]);
    wait_async0();        // my chunks landed
    __syncthreads();      // everyone's chunks landed
#pragma unroll
    for (int i = 0; i < 7; ++i) { xs[i] = si[t * 7 + i]; ys[i] = si[448 + t * 7 + i]; }
  } else
#endif
  {
    const int nc = nc0 + t;
    if (nc < nc_total) {
      const float* xp = x + nc * 7;
      const float* yp = y + nc * 7;
#pragma unroll
      for (int i = 0; i < 7; ++i) { xs[i] = xp[i]; ys[i] = yp[i]; }
    } else {
#pragma unroll
      for (int i = 0; i < 7; ++i) { xs[i] = 0.0f; ys[i] = 0.0f; }
    }
  }

  // 49 pairwise products, each reused ~10x across the ~470 CG terms.
  float p[49];
#pragma unroll
  for (int i = 0; i < 7; ++i)
#pragma unroll
    for (int j = 0; j < 7; ++j) p[i * 7 + j] = xs[i] * ys[j];

  float* row = &so[t * ROWF];
#pragma unroll
  for (int k4 = 0; k4 < NK; k4 += 4) {
    v4f ob;
#pragma unroll
    for (int u = 0; u < 4; ++u) {
      const int k = k4 + u;
      float a = 0.0f;
#pragma unroll
      for (int q = TT.koff[k]; q < TT.koff[k + 1]; ++q)
        a = fmaf(TT.coef[q], p[TT.prd[q]], a);   // coef/index fold to immediates
      ob[u] = a;
    }
    *(v4f*)&row[k4] = ob;                         // ds_store_b128, 16B-aligned
  }

  __syncthreads();

  // Transpose-writeback: consecutive lanes -> consecutive global float4s.
  // NK%4==0 so every float4 sits inside one LDS row, 16B aligned.
  const int base  = blockIdx.x * (NCT * NK);
  const int total = nc_total * NK;                // 249.6e6 < 2^31
#if USE_ASYNC
  if (full) {
    for (int q = t; q < (NCT * NK) / 4; q += NCT) {
      const int e = q * 4;
      const int r = e / NK;
      const int k = e - r * NK;
      async_l2g_b128_nt(out + base + e, &so[r * ROWF + k]);   // LDS -> global DMA
    }
    // s_endpgm performs an implicit wait-idle, draining ASYNCcnt before LDS is freed.
  } else
#endif
  {
    for (int q = t; q < (NCT * NK) / 4; q += NCT) {
      const int e = q * 4;
      const int r = e / NK;
      const int k = e - r * NK;
      v4f v = *(const v4f*)&so[r * ROWF + k];
      const int g = base + e;
      if (g < total)
        __builtin_nontemporal_store(v, (v4f*)(out + g));
    }
  }
}

// ---------------- launch ----------------
extern "C" void kernel_launch(void* const* d_in, const int* in_sizes, int n_in,
                              void* d_out, int out_size, void* d_ws, size_t ws_size,
                              hipStream_t stream) {
  const float* x = (const float*)d_in[0];
  const float* y = (const float*)d_in[1];
  float* out = (float*)d_out;
  const int nc_total = in_sizes[0] / 7;           // E*C = 1,600,000
  const int grid = (nc_total + NCT - 1) / NCT;    // 25,000 blocks of 64 threads
  tp_cg_kernel<<<grid, NCT, 0, stream>>>(x, y, out, nc_total);
}